// ActivationGATSingleHeadLayerIsotropic_83476984365548
// MI455X (gfx1250) — compile-verified
//
#include <hip/hip_runtime.h>
#include <hip/hip_bf16.h>

// GNN layer: segment-sum (scatter-add over edges) + BatchNorm(train) + ReLU.
// MI455X reasoning: h (51.2MB) and agg (51.2MB) both fit in the 192MB L2, so
// the 1.6GB gather + 1.6GB atomic-scatter traffic is L2-resident. Memory/atomic
// bound; WMMA used where natural: the BN column reduction as ones^T * X via
// V_WMMA_F32_16X16X4_F32 (native fp32 matrix op, no precision loss).

#define N_NODES 50000
#define N_EDGES 1600000
#define D_FEAT  256
#define BN_EPS  1e-5f

typedef float v2f __attribute__((ext_vector_type(2)));
typedef float v8f __attribute__((ext_vector_type(8)));

// d_ws float layout: [0..255]=colsum, [256..511]=colsumsq, [512..767]=scale, [768..1023]=bias

__device__ __forceinline__ void atomic_add_f32(float* p, float v) {
    __hip_atomic_fetch_add(p, v, __ATOMIC_RELAXED, __HIP_MEMORY_SCOPE_AGENT);
}

// ---- Kernel 1: zero the accumulation buffer (d_out) and the ws accumulators.
__global__ void k_zero(float4* __restrict__ out4, float* __restrict__ ws, long n4) {
    long i = (long)blockIdx.x * blockDim.x + threadIdx.x;
    long stride = (long)gridDim.x * blockDim.x;
    float4 z = make_float4(0.f, 0.f, 0.f, 0.f);
    for (; i < n4; i += stride) out4[i] = z;
    if (blockIdx.x == 0 && threadIdx.x < 512) ws[threadIdx.x] = 0.f;
}

// ---- Kernel 2: edge scatter-add. One wave32 per edge; each lane owns 8
// contiguous features (two b128 loads -> the wave covers the full 1KB row
// coalesced), then 8 hardware f32 atomics into out[dst].
__global__ __launch_bounds__(256) void k_scatter(const float* __restrict__ h,
                                                 const int* __restrict__ src,
                                                 const int* __restrict__ dst,
                                                 float* __restrict__ out) {
    const int lane = threadIdx.x & 31;
    const int waveInBlock = threadIdx.x >> 5;
    const int wavesPerBlock = blockDim.x >> 5;
    long wave = (long)blockIdx.x * wavesPerBlock + waveInBlock;
    long nWaves = (long)gridDim.x * wavesPerBlock;
    for (long e = wave; e < N_EDGES; e += nWaves) {
        const int s = src[e];            // uniform per wave -> scalarized
        const int d = dst[e];
        const float4* hrow = (const float4*)(h + (long)s * D_FEAT);
        float* orow = out + (long)d * D_FEAT;
        float4 a = hrow[lane * 2 + 0];   // features [8*lane .. 8*lane+3]
        float4 b = hrow[lane * 2 + 1];   // features [8*lane+4 .. 8*lane+7]
        const int base = lane * 8;
        atomic_add_f32(orow + base + 0, a.x);
        atomic_add_f32(orow + base + 1, a.y);
        atomic_add_f32(orow + base + 2, a.z);
        atomic_add_f32(orow + base + 3, a.w);
        atomic_add_f32(orow + base + 4, b.x);
        atomic_add_f32(orow + base + 5, b.y);
        atomic_add_f32(orow + base + 6, b.z);
        atomic_add_f32(orow + base + 7, b.w);
    }
}

// ---- Kernel 3: BN statistics via WMMA. One wave per block. Block (cg, rs):
// 16 features [cg*16 ..), 400 rows [rs*400 ..). Per iteration load a 4x16 f32
// tile of X (and X^2) into the B operand; A = all-ones 16x4, so
// D[m,n] = sum_k B[k,n] accumulates 4-row column sums on the matrix pipe.
// D VGPR0, lanes 0..15 = column N=lane. Atomically fold partials into ws.
__global__ __launch_bounds__(32) void k_stats(const float* __restrict__ x,
                                              float* __restrict__ ws) {
    const int lane = threadIdx.x;                 // 0..31, full wave active (EXEC all 1s)
    const int col  = blockIdx.x * 16 + (lane & 15);
    const int krow = (lane >> 4) * 2;             // lanes 0-15: K=0,1; lanes 16-31: K=2,3
    const int r0   = blockIdx.y * 400;

    v2f ones; ones[0] = 1.f; ones[1] = 1.f;
    v8f accS = {};   // column sums
    v8f accQ = {};   // column sums of squares

    for (int r = r0; r < r0 + 400; r += 4) {
        float e0 = x[(long)(r + krow + 0) * D_FEAT + col];
        float e1 = x[(long)(r + krow + 1) * D_FEAT + col];
        v2f b;  b[0]  = e0;      b[1]  = e1;
        v2f b2; b2[0] = e0 * e0; b2[1] = e1 * e1;
        accS = __builtin_amdgcn_wmma_f32_16x16x4_f32(false, ones, false, b,
                                                     (short)0, accS, false, false);
        accQ = __builtin_amdgcn_wmma_f32_16x16x4_f32(false, ones, false, b2,
                                                     (short)0, accQ, false, false);
    }
    if (lane < 16) {
        atomic_add_f32(&ws[col],       accS[0]);
        atomic_add_f32(&ws[256 + col], accQ[0]);
    }
}

// ---- Kernel 4: fold stats into per-feature scale/bias (1 block, 256 threads).
__global__ __launch_bounds__(256) void k_finalize(const float* __restrict__ gamma,
                                                  const float* __restrict__ beta,
                                                  float* __restrict__ ws) {
    const int c = threadIdx.x;
    const float inv_n = 1.f / (float)N_NODES;
    float mean = ws[c] * inv_n;
    float var  = ws[256 + c] * inv_n - mean * mean;
    float sc   = rsqrtf(var + BN_EPS) * gamma[c];
    ws[512 + c] = sc;
    ws[768 + c] = beta[c] - mean * sc;
}

// ---- Kernel 5: in-place normalize + ReLU, float4 vectorized.
__global__ __launch_bounds__(256) void k_norm(float4* __restrict__ out4,
                                              const float* __restrict__ ws, long n4) {
    long i = (long)blockIdx.x * blockDim.x + threadIdx.x;
    long stride = (long)gridDim.x * blockDim.x;
    const float* scale = ws + 512;
    const float* bias  = ws + 768;
    for (; i < n4; i += stride) {
        int c = (int)((i * 4) & (D_FEAT - 1));
        float4 v = out4[i];
        v.x = fmaxf(0.f, v.x * scale[c + 0] + bias[c + 0]);
        v.y = fmaxf(0.f, v.y * scale[c + 1] + bias[c + 1]);
        v.z = fmaxf(0.f, v.z * scale[c + 2] + bias[c + 2]);
        v.w = fmaxf(0.f, v.w * scale[c + 3] + bias[c + 3]);
        out4[i] = v;
    }
}

extern "C" void kernel_launch(void* const* d_in, const int* in_sizes, int n_in,
                              void* d_out, int out_size, void* d_ws, size_t ws_size,
                              hipStream_t stream) {
    const float* h     = (const float*)d_in[0];
    const float* gamma = (const float*)d_in[1];
    const float* beta  = (const float*)d_in[2];
    const int*   src   = (const int*)d_in[3];
    const int*   dst   = (const int*)d_in[4];
    float*  out  = (float*)d_out;
    float4* out4 = (float4*)d_out;
    float*  ws   = (float*)d_ws;

    const long n4 = (long)N_NODES * D_FEAT / 4;   // 3,200,000 float4

    k_zero<<<2048, 256, 0, stream>>>(out4, ws, n4);
    k_scatter<<<4096, 256, 0, stream>>>(h, src, dst, out);
    k_stats<<<dim3(D_FEAT / 16, N_NODES / 400), 32, 0, stream>>>(out, ws);
    k_finalize<<<1, 256, 0, stream>>>(gamma, beta, ws);
    k_norm<<<2048, 256, 0, stream>>>(out4, ws, n4);
}